// ParametricGTCNN_88974542504653
// MI455X (gfx1250) — compile-verified
//
#include <hip/hip_runtime.h>

typedef __attribute__((ext_vector_type(2))) float v2f;
typedef __attribute__((ext_vector_type(8))) float v8f;

#define Nn   2000
#define Tt   24
#define Bb   4
#define NT   48000        // Nn*Tt
#define HID  64
#define BNT  192000       // Bb*NT
#define BIG  12288000     // BNT*HID

// ---------------- utility ----------------
__global__ void k_zero(float* __restrict__ p, long n) {
    long i = (long)blockIdx.x * blockDim.x + threadIdx.x;
    if (i < n) p[i] = 0.0f;
}

// deg[r] += relu(s[group[e]])
__global__ void k_deg(const int* __restrict__ rows, const int* __restrict__ group,
                      const float* __restrict__ s_params, float* __restrict__ deg, int E) {
    int e = blockIdx.x * blockDim.x + threadIdx.x;
    if (e >= E) return;
    float s = s_params[group[e]];
    s = fmaxf(s, 0.0f);
    atomicAdd(&deg[rows[e]], s);
}

__global__ void k_rsqrt(const float* __restrict__ deg, float* __restrict__ dis, int n) {
    int i = blockIdx.x * blockDim.x + threadIdx.x;
    if (i >= n) return;
    float d = deg[i];
    dis[i] = (d > 0.0f) ? rsqrtf(d) : 0.0f;
}

__global__ void k_vhat(const int* __restrict__ rows, const int* __restrict__ cols,
                       const int* __restrict__ group, const float* __restrict__ s_params,
                       const float* __restrict__ dis, float* __restrict__ vhat, int E) {
    int e = blockIdx.x * blockDim.x + threadIdx.x;
    if (e >= E) return;
    float s = fmaxf(s_params[group[e]], 0.0f);
    vhat[e] = dis[rows[e]] * s * dis[cols[e]];
}

// x[b,0,n,t] -> H0[b*NT + t*N + n]
__global__ void k_permute(const float* __restrict__ x, float* __restrict__ H0) {
    int i = blockIdx.x * blockDim.x + threadIdx.x;   // over Bb*NT
    if (i >= BNT) return;
    int b = i / NT;
    int r = i - b * NT;
    int t = r / Nn;
    int n = r - t * Nn;
    H0[i] = x[((size_t)(b * Nn + n)) * Tt + t];
}

// scalar spmm: Y[b*NT + rows[e]] += vhat[e] * X[b*NT + cols[e]]  (Y pre-zeroed)
__global__ void k_spmm_s(const int* __restrict__ rows, const int* __restrict__ cols,
                         const float* __restrict__ vhat, const float* __restrict__ X,
                         float* __restrict__ Y, int E) {
    long i = (long)blockIdx.x * blockDim.x + threadIdx.x;   // over E*Bb
    if (i >= (long)E * Bb) return;
    int e = (int)(i % E);
    int b = (int)(i / E);
    atomicAdd(&Y[(size_t)b * NT + rows[e]], vhat[e] * X[(size_t)b * NT + cols[e]]);
}

// H1[node,o] = relu(H0[node]*W1[0,0,o] + P1[node]*W1[1,0,o] + P2[node]*W1[2,0,o] + b1[o])
__global__ void k_conv1(const float* __restrict__ H0, const float* __restrict__ P1,
                        const float* __restrict__ P2, const float* __restrict__ W1,
                        const float* __restrict__ b1, float* __restrict__ H1) {
    long i = (long)blockIdx.x * blockDim.x + threadIdx.x;   // over BNT*HID
    if (i >= (long)BNT * HID) return;
    int o = (int)(i & 63);
    long node = i >> 6;
    float v = H0[node] * W1[o] + P1[node] * W1[64 + o] + P2[node] * W1[128 + o] + b1[o];
    H1[i] = fmaxf(v, 0.0f);
}

// wide spmm: Y[b,rows[e],f] += vhat[e] * X[b,cols[e],f]   (Y pre-zeroed)
// block: 256 threads = 4 edges x 64 features ; blockIdx.y = batch
__global__ void k_spmm_w(const int* __restrict__ rows, const int* __restrict__ cols,
                         const float* __restrict__ vhat, const float* __restrict__ X,
                         float* __restrict__ Y, int E) {
    int e = blockIdx.x * 4 + (threadIdx.x >> 6);
    int f = threadIdx.x & 63;
    if (e >= E) return;
    int b = blockIdx.y;
    float w = vhat[e];
    size_t src = ((size_t)b * NT + cols[e]) * HID + f;
    size_t dst = ((size_t)b * NT + rows[e]) * HID + f;
    atomicAdd(&Y[dst], w * X[src]);
}

// O[m,0:64] (=|+=) X[m,0:64] @ W[64,64] (+bias) (relu) via V_WMMA_F32_16X16X4_F32
// block = 256 threads = 8 waves, each wave does a 16-row stripe; grid.x = M/128
// Compile-time epilogue flags -> branch-free stores, batched accumulate loads.
template <bool HAS_BIAS, bool ACCUM, bool RELU>
__global__ void k_gemm64(const float* __restrict__ X, const float* __restrict__ W,
                         const float* __restrict__ bias, float* __restrict__ O,
                         int M) {
    __shared__ float sW[64 * 64];
    for (int i = threadIdx.x; i < 64 * 64; i += blockDim.x) sW[i] = W[i];
    __syncthreads();

    int wave = threadIdx.x >> 5;             // 0..7
    int lane = threadIdx.x & 31;
    int row0 = (blockIdx.x * 8 + wave) * 16;
    if (row0 >= M) return;                   // uniform per wave (M % 128 == 0 here)
    int hi = lane >> 4;                      // 0 or 1
    int lo = lane & 15;

    const float* Xr = X + (size_t)(row0 + lo) * 64;   // this lane's A row
    const v2f* Xr2 = (const v2f*)Xr;

    // prefetch next block's stripe of X (streaming, known-ahead address)
    __builtin_prefetch(Xr + (size_t)128 * 64, 0, 3);

    for (int n0 = 0; n0 < 64; n0 += 16) {
        v8f acc = {};
#pragma unroll
        for (int k0 = 0; k0 < 64; k0 += 4) {
            // A frag: lane(L) holds A[L%16][2*(L/16)+j], j=0,1
            v2f a = Xr2[(k0 >> 1) + hi];
            // B frag: lane(L) holds B[2*(L/16)+j][L%16], j=0,1
            int ka = k0 + 2 * hi;
            v2f bfrag;
            bfrag.x = sW[ka * 64 + n0 + lo];
            bfrag.y = sW[(ka + 1) * 64 + n0 + lo];
            acc = __builtin_amdgcn_wmma_f32_16x16x4_f32(
                false, a, false, bfrag, (short)0, acc, false, false);
        }
        // D layout: VGPR v -> row m = v + 8*hi, col = n0 + lo
        float* Ob = O + (size_t)(row0 + 8 * hi) * 64 + n0 + lo;
        float prev[8];
        if (ACCUM) {
#pragma unroll
            for (int v = 0; v < 8; ++v) prev[v] = Ob[(size_t)v * 64];
        }
        float bval = HAS_BIAS ? bias[n0 + lo] : 0.0f;
#pragma unroll
        for (int v = 0; v < 8; ++v) {
            float val = acc[v];
            if (HAS_BIAS) val += bval;
            if (ACCUM)    val += prev[v];
            if (RELU)     val = fmaxf(val, 0.0f);
            Ob[(size_t)v * 64] = val;
        }
    }
}

// out[b,n] = head_b + (1/T) * sum_t sum_h H2[b, t*N+n, h] * head_w[h]
__global__ void k_pool_head(const float* __restrict__ H2, const float* __restrict__ hw,
                            const float* __restrict__ hb, float* __restrict__ out) {
    int i = blockIdx.x * blockDim.x + threadIdx.x;   // over Bb*Nn
    if (i >= Bb * Nn) return;
    int b = i / Nn;
    int n = i - b * Nn;
    float acc = 0.0f;
    for (int t = 0; t < Tt; ++t) {
        const float* row = H2 + ((size_t)b * NT + (size_t)t * Nn + n) * HID;
#pragma unroll
        for (int h = 0; h < HID; ++h) acc += row[h] * hw[h];
    }
    out[i] = acc * (1.0f / Tt) + hb[0];
}

extern "C" void kernel_launch(void* const* d_in, const int* in_sizes, int n_in,
                              void* d_out, int out_size, void* d_ws, size_t ws_size,
                              hipStream_t stream) {
    const float* x        = (const float*)d_in[0];
    const float* s_params = (const float*)d_in[1];
    const float* W1       = (const float*)d_in[2];
    const float* b1       = (const float*)d_in[3];
    const float* W2       = (const float*)d_in[4];
    const float* b2       = (const float*)d_in[5];
    const float* head_w   = (const float*)d_in[6];
    const float* head_b   = (const float*)d_in[7];
    const int*   rows     = (const int*)d_in[8];
    const int*   cols     = (const int*)d_in[9];
    const int*   group    = (const int*)d_in[10];
    const int    E        = in_sizes[8];

    float* ws   = (float*)d_ws;
    float* deg  = ws;              // NT
    float* dis  = deg  + NT;       // NT
    float* vhat = dis  + NT;       // E
    float* H0   = vhat + E;        // BNT
    float* P1   = H0   + BNT;      // BNT
    float* P2   = P1   + BNT;      // BNT
    float* bufA = P2   + BNT;      // BIG  (H1, later Q2)
    float* bufB = bufA + BIG;      // BIG  (Q1)
    float* bufC = bufB + BIG;      // BIG  (O / H2)

    const int TPB = 256;
    auto blks = [](long n) { return (unsigned)((n + 255) / 256); };

    // edge weights / normalization
    k_zero<<<blks(NT), TPB, 0, stream>>>(deg, NT);
    k_deg<<<blks(E), TPB, 0, stream>>>(rows, group, s_params, deg, E);
    k_rsqrt<<<blks(NT), TPB, 0, stream>>>(deg, dis, NT);
    k_vhat<<<blks(E), TPB, 0, stream>>>(rows, cols, group, s_params, dis, vhat, E);

    // layer 1 (F_IN = 1 -> scalar field)
    k_permute<<<blks(BNT), TPB, 0, stream>>>(x, H0);
    k_zero<<<blks(BNT), TPB, 0, stream>>>(P1, BNT);
    k_zero<<<blks(BNT), TPB, 0, stream>>>(P2, BNT);
    k_spmm_s<<<blks((long)E * Bb), TPB, 0, stream>>>(rows, cols, vhat, H0, P1, E);
    k_spmm_s<<<blks((long)E * Bb), TPB, 0, stream>>>(rows, cols, vhat, P1, P2, E);
    k_conv1<<<blks((long)BNT * HID), TPB, 0, stream>>>(H0, P1, P2, W1, b1, bufA);

    // layer 2: O = H1@W2[0] + b2 ; Q1 = A H1 ; O += Q1@W2[1] ; Q2 = A Q1 ; O += Q2@W2[2] ; relu
    const unsigned gemm_grid = BNT / 128;    // 192000 / 128 = 1500, exact
    dim3 sg((unsigned)((E + 3) / 4), Bb, 1);

    k_gemm64<true, false, false><<<gemm_grid, TPB, 0, stream>>>(bufA, W2 + 0 * 4096, b2, bufC, BNT);
    k_zero<<<blks(BIG), TPB, 0, stream>>>(bufB, BIG);
    k_spmm_w<<<sg, TPB, 0, stream>>>(rows, cols, vhat, bufA, bufB, E);
    k_gemm64<false, true, false><<<gemm_grid, TPB, 0, stream>>>(bufB, W2 + 1 * 4096, nullptr, bufC, BNT);
    k_zero<<<blks(BIG), TPB, 0, stream>>>(bufA, BIG);
    k_spmm_w<<<sg, TPB, 0, stream>>>(rows, cols, vhat, bufB, bufA, E);
    k_gemm64<false, true, true><<<gemm_grid, TPB, 0, stream>>>(bufA, W2 + 2 * 4096, nullptr, bufC, BNT);

    // pool over time + head
    k_pool_head<<<blks(Bb * Nn), TPB, 0, stream>>>(bufC, head_w, head_b, (float*)d_out);
}